// IIRFilter_30305289240911
// MI455X (gfx1250) — compile-verified
//
#include <hip/hip_runtime.h>

typedef __attribute__((ext_vector_type(2))) float v2f;
typedef __attribute__((ext_vector_type(4))) float v4f;
typedef __attribute__((ext_vector_type(8))) float v8f;

#define NB 16       // batch
#define NT 32768    // time length
#define NF 30       // filters
#define NK 7        // filter taps
#define ORD 6       // IIR order = NK-1
#define CH 16       // chunk (M, and K for the Toeplitz part)
#define LDT 16      // Tc row stride (floats)
#define LDG 12      // Gc row stride (floats)
#define FSTRIDE 512 // floats of workspace per filter

// -----------------------------------------------------------------------------
// Setup: per-filter chunk matrices.
//   Tc[t][j] = hc[t-j]  (t>=j)  combined impulse response (FIR b ∘ IIR 1/a)
//   Gc[t][k]   k=0..5 :  (A^{t+1})[0][k]     (carry of y[-1-k])
//   Gc[t][6+k] k=0..5 :  Σ_{m=k+1..6} b[m] h[t+1+k-m]   (carry of x[-1-k])
// -----------------------------------------------------------------------------
__global__ void iir_setup_kernel(const float* __restrict__ bs,
                                 const float* __restrict__ as_,
                                 float* __restrict__ ws) {
  int f = threadIdx.x;
  if (f >= NF) return;
  float a[NK], b[NK];
  float a0 = as_[f * NK];
  for (int k = 0; k < NK; ++k) {
    a[k] = as_[f * NK + k] / a0;
    b[k] = bs[f * NK + k] / a0;
  }
  // IIR-only impulse response h
  float h[CH];
  for (int t = 0; t < CH; ++t) {
    float v = (t == 0) ? 1.0f : 0.0f;
    for (int k = 1; k < NK; ++k)
      if (k <= t) v -= a[k] * h[t - k];
    h[t] = v;
  }
  // Combined impulse response hc = b * h
  float hc[CH];
  for (int n = 0; n < CH; ++n) {
    float v = 0.0f;
    for (int m = 0; m < NK; ++m)
      if (m <= n) v += b[m] * h[n - m];
    hc[n] = v;
  }
  float* Tc = ws + f * FSTRIDE;
  float* Gc = Tc + 256;
  for (int t = 0; t < CH; ++t)
    for (int j = 0; j < CH; ++j)
      Tc[t * LDT + j] = (t >= j) ? hc[t - j] : 0.0f;
  // G: row0 of companion-matrix powers A^{t+1}
  float r[ORD];
  for (int k = 0; k < ORD; ++k) r[k] = -a[k + 1];
  for (int t = 0; t < CH; ++t) {
    for (int k = 0; k < ORD; ++k) Gc[t * LDG + k] = r[k];
    float rn[ORD];
    for (int j = 0; j < ORD; ++j) {
      float v = -r[0] * a[j + 1];
      if (j < ORD - 1) v += r[j + 1];
      rn[j] = v;
    }
    for (int k = 0; k < ORD; ++k) r[k] = rn[k];
  }
  // Hx: carry of pre-chunk x values through the FIR memory
  for (int t = 0; t < CH; ++t)
    for (int k = 0; k < ORD; ++k) {
      float v = 0.0f;
      for (int m = k + 1; m < NK; ++m) {
        int idx = t + 1 + k - m;
        if (idx >= 0) v += b[m] * h[idx];
      }
      Gc[t * LDG + ORD + k] = v;
    }
}

// -----------------------------------------------------------------------------
// Main: one wave (32 lanes) per filter; 16 batches ride the WMMA N dimension.
// Per 16-step chunk: 7 x V_WMMA_F32_16X16X4_F32 in TWO independent chains:
//   accT (4 WMMAs, Toeplitz x input tile)  -- off the recurrence cycle
//   accS (3 WMMAs, state carry)            -- the loop-carried cycle
// so the serial dependence is ~3 WMMA + v_add + ds_bpermute per 16 samples.
// -----------------------------------------------------------------------------
__global__ void __launch_bounds__(32)
iir_wmma_kernel(const float* __restrict__ x,
                const float* __restrict__ ws,
                float* __restrict__ out) {
  const int f = blockIdx.x;
  const int lane = threadIdx.x & 31;
  const int n = lane & 15;          // batch column (WMMA N)
  const bool hi = lane >= 16;
  const int koff = hi ? 2 : 0;      // K sub-row pair held by this lane half

  const float* Tc = ws + f * FSTRIDE;
  const float* Gc = Tc + 256;

  // Constant A fragments (A-matrix layout: V0 = K 0/2, V1 = K 1/3 per half)
  v2f TA[4], GA[3];
#pragma unroll
  for (int q = 0; q < 4; ++q)
    TA[q] = *(const v2f*)(Tc + n * LDT + 4 * q + koff);
#pragma unroll
  for (int q = 0; q < 3; ++q)
    GA[q] = *(const v2f*)(Gc + n * LDG + 4 * q + koff);

  const float* xrow = x + (size_t)n * NT;           // batch row of input
  float* orow = out + ((size_t)n * NF + f) * NT;    // (batch, filter) output row

  // State B fragments, zero initial conditions
  v2f Bs0 = {0.f, 0.f}, Bs1 = {0.f, 0.f}, Bs2 = {0.f, 0.f};

  for (int t0 = 0; t0 < NT; t0 += CH) {
    // x-tile B fragments: XB[q] = { x[n][t0+4q+koff], x[n][t0+4q+koff+1] }
    v2f XB[4];
#pragma unroll
    for (int q = 0; q < 4; ++q)
      XB[q] = *(const v2f*)(xrow + t0 + 4 * q + koff);

    if (t0 + CH < NT)
      __builtin_prefetch(xrow + t0 + CH, 0, 3);   // keep next tile near

    // Chain 1: Toeplitz part (independent of loop-carried state)
    v8f accT = {};
#pragma unroll
    for (int q = 0; q < 4; ++q)
      accT = __builtin_amdgcn_wmma_f32_16x16x4_f32(
          false, TA[q], false, XB[q], (short)0, accT, false, false);

    // Chain 2: state carry (the recurrence cycle)
    v8f accS = {};
    accS = __builtin_amdgcn_wmma_f32_16x16x4_f32(
        false, GA[0], false, Bs0, (short)0, accS, false, false);
    accS = __builtin_amdgcn_wmma_f32_16x16x4_f32(
        false, GA[1], false, Bs1, (short)0, accS, false, false);
    accS = __builtin_amdgcn_wmma_f32_16x16x4_f32(
        false, GA[2], false, Bs2, (short)0, accS, false, false);

    v8f acc = accT + accS;   // 8 dual-issuable v_add_f32

    // D layout: VGPR r = time row r (lanes 0-15) / r+8 (lanes 16-31), col = n
    float* po = orow + t0 + (hi ? 8 : 0);
    *(v4f*)po       = v4f{acc[0], acc[1], acc[2], acc[3]};
    *(v4f*)(po + 4) = v4f{acc[4], acc[5], acc[6], acc[7]};

    // Next-chunk state, built in registers.
    // Broadcast hi-half rows (time 10..15) to all lanes:
    const int src = n + 16;
    float y15 = __shfl(acc[7], src);
    float y14 = __shfl(acc[6], src);
    float y13 = __shfl(acc[5], src);
    float y12 = __shfl(acc[4], src);
    float y11 = __shfl(acc[3], src);
    float y10 = __shfl(acc[2], src);
    // S = (y[-1..-6], x[-1..-6]); B layout per lane half as with A.
    Bs0 = hi ? v2f{y13, y12} : v2f{y15, y14};
    Bs1 = hi ? v2f{XB[3].y, XB[3].x} : v2f{y11, y10};   // hi: x[t0+15], x[t0+14]
    Bs2 = hi ? v2f{XB[2].y, XB[2].x}                    // hi: x[t0+11], x[t0+10]
             : v2f{XB[3].y, XB[3].x};                   // lo: x[t0+13], x[t0+12]
  }
}

extern "C" void kernel_launch(void* const* d_in, const int* in_sizes, int n_in,
                              void* d_out, int out_size, void* d_ws, size_t ws_size,
                              hipStream_t stream) {
  const float* x   = (const float*)d_in[0];   // (16, 32768) f32
  const float* bs  = (const float*)d_in[1];   // (30, 7) f32
  const float* as_ = (const float*)d_in[2];   // (30, 7) f32
  float* out = (float*)d_out;                 // (16, 30, 32768) f32
  float* ws  = (float*)d_ws;                  // >= 30*512*4 bytes used

  iir_setup_kernel<<<1, 32, 0, stream>>>(bs, as_, ws);
  iir_wmma_kernel<<<NF, 32, 0, stream>>>(x, ws, out);
}